// proj_layer_56478819942623
// MI455X (gfx1250) — compile-verified
//
#include <hip/hip_runtime.h>
#include <math.h>

typedef __attribute__((ext_vector_type(2))) float v2f;
typedef __attribute__((ext_vector_type(8))) float v8f;

#define RD 192
#define RH 192
#define WW 128
#define PROJ 4
#define VOL_STRIDE (128*128*128)
#define TILES_PER_P (RD * (RH/16))   // 192*12 = 2304

__device__ __forceinline__ int clamp127(int v) {
    return v < 0 ? 0 : (v > 127 ? 127 : v);
}

// One coordinate -> trilinear samples from both batch volumes (shared idx/weights).
__device__ __forceinline__ void trilinear2(const float* __restrict__ v0,
                                           const float* __restrict__ v1,
                                           float cx, float cy, float cz,
                                           float& r0, float& r1) {
    // coords order from grids: c0 -> ix (W axis), c1 -> iy (H), c2 -> iz (D)
    float ix = (cx + 1.0f) * 63.5f;
    float iy = (cy + 1.0f) * 63.5f;
    float iz = (cz + 1.0f) * 63.5f;
    float xf = floorf(ix), yf = floorf(iy), zf = floorf(iz);
    float fx = ix - xf, fy = iy - yf, fz = iz - zf;
    int x0 = (int)xf, y0 = (int)yf, z0 = (int)zf;
    // per-axis weights, zeroed when the corner index is out of range (matches
    // the reference's per-corner `valid` mask since the product of any zeroed
    // axis weight zeroes that corner's contribution)
    float wx0 = (x0 >=  0 && x0 <= 127) ? (1.0f - fx) : 0.0f;
    float wx1 = (x0 >= -1 && x0 <= 126) ? fx          : 0.0f;
    float wy0 = (y0 >=  0 && y0 <= 127) ? (1.0f - fy) : 0.0f;
    float wy1 = (y0 >= -1 && y0 <= 126) ? fy          : 0.0f;
    float wz0 = (z0 >=  0 && z0 <= 127) ? (1.0f - fz) : 0.0f;
    float wz1 = (z0 >= -1 && z0 <= 126) ? fz          : 0.0f;
    int xi0 = clamp127(x0),     xi1 = clamp127(x0 + 1);
    int yi0 = clamp127(y0),     yi1 = clamp127(y0 + 1);
    int zi0 = clamp127(z0),     zi1 = clamp127(z0 + 1);
    // idx = (zi*H + yi)*W + xi
    int b00 = (zi0*128 + yi0)*128;
    int b01 = (zi0*128 + yi1)*128;
    int b10 = (zi1*128 + yi0)*128;
    int b11 = (zi1*128 + yi1)*128;
    float w000 = wz0*wy0*wx0, w001 = wz0*wy0*wx1;
    float w010 = wz0*wy1*wx0, w011 = wz0*wy1*wx1;
    float w100 = wz1*wy0*wx0, w101 = wz1*wy0*wx1;
    float w110 = wz1*wy1*wx0, w111 = wz1*wy1*wx1;
    r0 = w000*v0[b00+xi0] + w001*v0[b00+xi1]
       + w010*v0[b01+xi0] + w011*v0[b01+xi1]
       + w100*v0[b10+xi0] + w101*v0[b10+xi1]
       + w110*v0[b11+xi0] + w111*v0[b11+xi1];
    r1 = w000*v1[b00+xi0] + w001*v1[b00+xi1]
       + w010*v1[b01+xi0] + w011*v1[b01+xi1]
       + w100*v1[b10+xi0] + w101*v1[b10+xi1]
       + w110*v1[b11+xi0] + w111*v1[b11+xi1];
}

// One wave per 16-row tile of (p, rd). Lane L (L<16) owns row L with w-chunk
// slots {0,1} = w in [0,32)/[32,64); lane L+16 owns row L with slots {2,3} =
// [64,96)/[96,128). This matches the V_WMMA_F32_16X16X4_F32 A-matrix VGPR
// layout exactly, so a single WMMA against an all-ones B produces the fp32
// row sums (replicated across columns) on the matrix pipe.
__global__ void __launch_bounds__(256) proj_ray_wmma_kernel(
        const float* __restrict__ x,
        const float* __restrict__ grids,
        const float* __restrict__ dx,
        float* __restrict__ proj) {
    const int lane = threadIdx.x & 31;
    const int wave = threadIdx.x >> 5;
    const int tile = blockIdx.x * 8 + wave;          // 1152*8 = 9216 tiles, exact
    const int p    = tile / TILES_PER_P;
    int rem        = tile % TILES_PER_P;
    const int rd   = rem / (RH/16);
    const int rh0  = (rem % (RH/16)) * 16;
    const int row  = lane & 15;
    const int half = lane >> 4;

    const float* g  = grids + (size_t)(((p*RD + rd)*RH + (rh0 + row)) * WW) * 3;
    const float* v0 = x;
    const float* v1 = x + VOL_STRIDE;
    const int w0 = half * 64;        // first w-chunk base for this lane
    const int w1 = half * 64 + 32;   // second w-chunk base

    float a00 = 0.0f, a01 = 0.0f;    // batch-0 partials (A-matrix K-slot pair)
    float a10 = 0.0f, a11 = 0.0f;    // batch-1 partials

    for (int t = 0; t < 32; ++t) {
        const float* gp0 = g + (size_t)(w0 + t) * 3;
        const float* gp1 = g + (size_t)(w1 + t) * 3;
        // run the grid stream ahead (global_prefetch_b8, speculative)
        __builtin_prefetch(gp0 + 48, 0, 1);
        __builtin_prefetch(gp1 + 48, 0, 1);
        // grids are read-once: non-temporal, keep L2 for the volume
        float c0x = __builtin_nontemporal_load(gp0 + 0);
        float c0y = __builtin_nontemporal_load(gp0 + 1);
        float c0z = __builtin_nontemporal_load(gp0 + 2);
        float c1x = __builtin_nontemporal_load(gp1 + 0);
        float c1y = __builtin_nontemporal_load(gp1 + 1);
        float c1z = __builtin_nontemporal_load(gp1 + 2);
        float r0, r1;
        trilinear2(v0, v1, c0x, c0y, c0z, r0, r1);
        a00 += r0; a10 += r1;
        trilinear2(v0, v1, c1x, c1y, c1z, r0, r1);
        a01 += r0; a11 += r1;
    }

    v2f A0 = {a00, a01};
    v2f A1 = {a10, a11};
    v2f Bv = {1.0f, 1.0f};
    v8f c0 = {0.0f, 0.0f, 0.0f, 0.0f, 0.0f, 0.0f, 0.0f, 0.0f};
    v8f c1 = {0.0f, 0.0f, 0.0f, 0.0f, 0.0f, 0.0f, 0.0f, 0.0f};
#if __has_builtin(__builtin_amdgcn_wmma_f32_16x16x4_f32)
    c0 = __builtin_amdgcn_wmma_f32_16x16x4_f32(false, A0, false, Bv, (short)0, c0, false, false);
    c1 = __builtin_amdgcn_wmma_f32_16x16x4_f32(false, A1, false, Bv, (short)0, c1, false, false);
#else
    asm volatile("v_wmma_f32_16x16x4_f32 %0, %1, %2, %0" : "+v"(c0) : "v"(A0), "v"(Bv));
    asm volatile("v_wmma_f32_16x16x4_f32 %0, %1, %2, %0" : "+v"(c1) : "v"(A1), "v"(Bv));
#endif

    // D layout: lane 0 holds rows 0..7 in c[0..7] (any column), lane 16 rows 8..15.
    if (row == 0) {
        const int rbase = half * 8;
        const float* dxr = dx + ((size_t)(p*RD + rd)*RH) + rh0 + rbase;
        float* p0 = proj + ((size_t)(0*PROJ + p)*RD + rd)*RH + rh0 + rbase;
        float* p1 = proj + ((size_t)(1*PROJ + p)*RD + rd)*RH + rh0 + rbase;
        #pragma unroll
        for (int j = 0; j < 8; ++j) {
            float dd = dxr[j];
            p0[j] = c0[j] * dd;
            p1[j] = c1[j] * dd;
        }
    }
}

// Nearest-neighbor 192x192 -> 256x256 per (b, p) image.
__global__ void __launch_bounds__(256) resize_nearest_kernel(
        const float* __restrict__ proj, float* __restrict__ out) {
    int tid = blockIdx.x * blockDim.x + threadIdx.x;   // 2*4*256*256 threads
    int oj = tid & 255;
    int oi = (tid >> 8) & 255;
    int c  = tid >> 16;                                // b*4 + p
    int hi = (oi * 3) >> 2;                            // == min(oi*192//256, 191)
    int wi = (oj * 3) >> 2;
    out[tid] = proj[(size_t)c * (RD*RH) + hi*RH + wi];
}

extern "C" void kernel_launch(void* const* d_in, const int* in_sizes, int n_in,
                              void* d_out, int out_size, void* d_ws, size_t ws_size,
                              hipStream_t stream) {
    (void)in_sizes; (void)n_in; (void)out_size; (void)ws_size;
    const float* x     = (const float*)d_in[0];   // (2,128,128,128) f32
    const float* grids = (const float*)d_in[1];   // (4,192,192,128,3) f32
    const float* dx    = (const float*)d_in[2];   // (4,192,192) f32
    float* out  = (float*)d_out;                  // (2,4,256,256) f32
    float* proj = (float*)d_ws;                   // scratch: 2*4*192*192 f32 = 1.18 MB

    proj_ray_wmma_kernel<<<1152, 256, 0, stream>>>(x, grids, dx, proj);
    resize_nearest_kernel<<<2048, 256, 0, stream>>>(proj, out);
}